// LossFunction_85873576116767
// MI455X (gfx1250) — compile-verified
//
#include <hip/hip_runtime.h>
#include <math.h>

// ---------------------------------------------------------------------------
// Angular-prototypical-style hard-mining loss over the 8192x8192 cosine Gram
// matrix, computed in two streaming WMMA passes (never materializing sim).
// fp32 path: v_wmma_f32_16x16x4_f32 (exact enough for exp(50*(sim-0.5))).
// ---------------------------------------------------------------------------

typedef __attribute__((ext_vector_type(2))) float v2f;
typedef __attribute__((ext_vector_type(8))) float v8f;

#define TILE    16
#define DDIM    256
#define LDA     260    // padded LDS row stride (260 % 64 == 4) -> conflict-free A reads
#define NWAVES  8
#define THRESH  0.5f
#define MARGIN  0.1f
#define EPSF    1e-5f

// --------------------------- Pass 1: min_pos / max_neg ----------------------
__global__ __launch_bounds__(256)
void pass1_minmax(const float* __restrict__ feats, const int* __restrict__ labels,
                  float* __restrict__ minPos, float* __restrict__ maxNeg,
                  int nColTiles) {
  __shared__ float sA[TILE * LDA];
  __shared__ int   sLab[TILE];
  __shared__ float sRMin[NWAVES][TILE];
  __shared__ float sRMax[NWAVES][TILE];

  const int rowBase = blockIdx.x * TILE;
  const int tid = threadIdx.x;

  // Stage 16x256 A tile into LDS (coalesced float4 loads, padded stride).
  {
    const int r = tid >> 4;
    const int c = (tid & 15) * 16;
    const float4* src = (const float4*)(feats + (rowBase + r) * DDIM + c);
    float4* dst = (float4*)(&sA[r * LDA + c]);
    dst[0] = src[0]; dst[1] = src[1]; dst[2] = src[2]; dst[3] = src[3];
  }
  if (tid < TILE) sLab[tid] = labels[rowBase + tid];
  __syncthreads();

  const int lane  = tid & 31;
  const int wave  = tid >> 5;
  const int nIdx  = lane & 15;   // column within tile (B/C/D lane index)
  const int half  = lane >> 4;   // which half-wave
  const int khalf = half * 2;    // K sub-offset for A/B fragments
  const int mLane = nIdx;        // A-matrix row for this lane

  float minp[8], maxn[8];
  int rowLab[8];
  #pragma unroll
  for (int v = 0; v < 8; ++v) {
    minp[v]   =  __builtin_inff();
    maxn[v]   = -__builtin_inff();
    rowLab[v] = sLab[v + 8 * half];
  }

  for (int ct = wave; ct < nColTiles; ct += NWAVES) {
    const int c = ct * TILE + nIdx;
    const int labC = labels[c];
    const float* gB = feats + c * DDIM;

    v8f acc = {};
    #pragma unroll 8
    for (int k0 = 0; k0 < DDIM; k0 += 4) {
      v2f a = *(const v2f*)(&sA[mLane * LDA + k0 + khalf]);
      v2f b = *(const v2f*)(gB + k0 + khalf);
      acc = __builtin_amdgcn_wmma_f32_16x16x4_f32(false, a, false, b,
                                                  (short)0, acc, false, false);
    }

    #pragma unroll
    for (int v = 0; v < 8; ++v) {
      const float s = acc[v];
      if (rowLab[v] == labC) {
        if (s < 1.0f - EPSF) minp[v] = fminf(minp[v], s);   // drop self-sim ~1
      } else {
        maxn[v] = fmaxf(maxn[v], s);
      }
    }
  }

  // Butterfly reduce across the 16 lanes of each half-wave (deterministic).
  #pragma unroll
  for (int v = 0; v < 8; ++v) {
    #pragma unroll
    for (int m = 1; m < 16; m <<= 1) {
      minp[v] = fminf(minp[v], __shfl_xor(minp[v], m, 32));
      maxn[v] = fmaxf(maxn[v], __shfl_xor(maxn[v], m, 32));
    }
  }
  if (nIdx == 0) {
    #pragma unroll
    for (int v = 0; v < 8; ++v) {
      sRMin[wave][v + 8 * half] = minp[v];
      sRMax[wave][v + 8 * half] = maxn[v];
    }
  }
  __syncthreads();
  if (tid < TILE) {
    float mn =  __builtin_inff();
    float mx = -__builtin_inff();
    #pragma unroll
    for (int w = 0; w < NWAVES; ++w) {
      mn = fminf(mn, sRMin[w][tid]);
      mx = fmaxf(mx, sRMax[w][tid]);
    }
    minPos[rowBase + tid] = mn;
    maxNeg[rowBase + tid] = mx;
  }
}

// ------------------- Pass 2: masked exp-sums with re-masking ----------------
__global__ __launch_bounds__(256)
void pass2_sums(const float* __restrict__ feats, const int* __restrict__ labels,
                const float* __restrict__ minPos, const float* __restrict__ maxNeg,
                float* __restrict__ posSum, float* __restrict__ negSum,
                int nColTiles) {
  __shared__ float sA[TILE * LDA];
  __shared__ int   sLab[TILE];
  __shared__ float sThrN[TILE];   // min_pos - MARGIN  (neg_sel: sim > thrN)
  __shared__ float sThrP[TILE];   // max_neg + MARGIN  (pos_sel: sim < thrP)
  __shared__ float sRP[NWAVES][TILE];
  __shared__ float sRN[NWAVES][TILE];

  const int rowBase = blockIdx.x * TILE;
  const int tid = threadIdx.x;

  {
    const int r = tid >> 4;
    const int c = (tid & 15) * 16;
    const float4* src = (const float4*)(feats + (rowBase + r) * DDIM + c);
    float4* dst = (float4*)(&sA[r * LDA + c]);
    dst[0] = src[0]; dst[1] = src[1]; dst[2] = src[2]; dst[3] = src[3];
  }
  if (tid < TILE) {
    sLab[tid]  = labels[rowBase + tid];
    sThrN[tid] = minPos[rowBase + tid] - MARGIN;  // inf stays inf -> selects none
    sThrP[tid] = maxNeg[rowBase + tid] + MARGIN;  // -inf stays -inf -> selects none
  }
  __syncthreads();

  const int lane  = tid & 31;
  const int wave  = tid >> 5;
  const int nIdx  = lane & 15;
  const int half  = lane >> 4;
  const int khalf = half * 2;
  const int mLane = nIdx;

  float ps[8], ns[8], thrN[8], thrP[8];
  int rowLab[8];
  #pragma unroll
  for (int v = 0; v < 8; ++v) {
    ps[v] = 0.0f; ns[v] = 0.0f;
    rowLab[v] = sLab[v + 8 * half];
    thrN[v]   = sThrN[v + 8 * half];
    thrP[v]   = sThrP[v + 8 * half];
  }

  for (int ct = wave; ct < nColTiles; ct += NWAVES) {
    const int c = ct * TILE + nIdx;
    const int labC = labels[c];
    const float* gB = feats + c * DDIM;

    v8f acc = {};
    #pragma unroll 8
    for (int k0 = 0; k0 < DDIM; k0 += 4) {
      v2f a = *(const v2f*)(&sA[mLane * LDA + k0 + khalf]);
      v2f b = *(const v2f*)(gB + k0 + khalf);
      acc = __builtin_amdgcn_wmma_f32_16x16x4_f32(false, a, false, b,
                                                  (short)0, acc, false, false);
    }

    #pragma unroll
    for (int v = 0; v < 8; ++v) {
      const float s = acc[v];
      if (rowLab[v] == labC) {
        if (s < 1.0f - EPSF && s < thrP[v])
          ps[v] += expf(-2.0f * (s - THRESH));
      } else {
        if (s > thrN[v])
          ns[v] += expf(50.0f * (s - THRESH));
      }
    }
  }

  #pragma unroll
  for (int v = 0; v < 8; ++v) {
    #pragma unroll
    for (int m = 1; m < 16; m <<= 1) {
      ps[v] += __shfl_xor(ps[v], m, 32);
      ns[v] += __shfl_xor(ns[v], m, 32);
    }
  }
  if (nIdx == 0) {
    #pragma unroll
    for (int v = 0; v < 8; ++v) {
      sRP[wave][v + 8 * half] = ps[v];
      sRN[wave][v + 8 * half] = ns[v];
    }
  }
  __syncthreads();
  if (tid < TILE) {
    float p = 0.0f, n = 0.0f;
    #pragma unroll
    for (int w = 0; w < NWAVES; ++w) { p += sRP[w][tid]; n += sRN[w][tid]; }
    posSum[rowBase + tid] = p;
    negSum[rowBase + tid] = n;
  }
}

// --------------------------- Pass 3: final scalars --------------------------
__global__ __launch_bounds__(256)
void pass3_final(const float* __restrict__ posSum, const float* __restrict__ negSum,
                 float* __restrict__ out, int B) {
  __shared__ float sL[256];
  __shared__ float sV[256];
  const int tid = threadIdx.x;

  float ls = 0.0f, vc = 0.0f;
  for (int r = tid; r < B; r += 256) {
    const float p = posSum[r];
    const float n = negSum[r];
    const bool valid = (p > 0.0f) && (n > 0.0f);   // exp terms are always > 0
    if (valid) {
      ls += 0.5f * log1pf(p) + 0.02f * log1pf(n);  // 1/SCALE_POS, 1/SCALE_NEG
      vc += 1.0f;
    }
  }
  sL[tid] = ls; sV[tid] = vc;
  __syncthreads();
  for (int s = 128; s > 0; s >>= 1) {
    if (tid < s) { sL[tid] += sL[tid + s]; sV[tid] += sV[tid + s]; }
    __syncthreads();
  }
  if (tid == 0) {
    out[0] = sL[0] / (float)B;           // loss
    out[1] = 1.0f - sV[0] / (float)B;    // prec1 = mean(1 - valid)
  }
}

// ---------------------------------------------------------------------------
extern "C" void kernel_launch(void* const* d_in, const int* in_sizes, int n_in,
                              void* d_out, int out_size, void* d_ws, size_t ws_size,
                              hipStream_t stream) {
  const float* feats  = (const float*)d_in[0];
  const int*   labels = (const int*)d_in[1];
  const int B = in_sizes[1];            // 8192 rows (D fixed at 256)
  const int nTiles = B / TILE;          // 512 row/col tiles

  float* minPos = (float*)d_ws;
  float* maxNeg = minPos + B;
  float* posSum = maxNeg + B;
  float* negSum = posSum + B;
  float* out    = (float*)d_out;

  pass1_minmax<<<nTiles, 256, 0, stream>>>(feats, labels, minPos, maxNeg, nTiles);
  pass2_sums  <<<nTiles, 256, 0, stream>>>(feats, labels, minPos, maxNeg,
                                           posSum, negSum, nTiles);
  pass3_final <<<1, 256, 0, stream>>>(posSum, negSum, out, B);
}